// ScatterDataReadout_56195352101149
// MI455X (gfx1250) — compile-verified
//
#include <hip/hip_runtime.h>

#define DIM 128
#define HIDDEN 384
#define BB 4
#define N_DATA 100000
#define EDGES 200000
#define N_LOGICAL 4096
#define EPSF 1e-5f
#define MT 4  // 16-row WMMA tiles per block -> 64 tokens/block

typedef __attribute__((ext_vector_type(2))) float v2f;
typedef __attribute__((ext_vector_type(8))) float v8f;

static __device__ inline v8f wmma4(v2f a, v2f b, v8f c) {
  // D = A(16x4 f32) * B(4x16 f32) + C(16x16 f32)
  return __builtin_amdgcn_wmma_f32_16x16x4_f32(
      false, a, false, b, (short)0, c, false, false);
}

static __device__ inline float fast_sigmoid(float x) {
  // 1/(1+e^-x): v_exp_f32 + single v_rcp_f32 (hardware approx, ~1 ulp)
  return __builtin_amdgcn_rcpf(1.0f + __expf(-x));
}

static __device__ inline float fast_tanh(float x) {
  // tanh(x) = 1 - 2/(e^(2x)+1); correct limits at +/-inf
  return 1.0f - 2.0f * __builtin_amdgcn_rcpf(__expf(2.0f * x) + 1.0f);
}

// ---------------------------------------------------------------------------
// Gated FFN on 64-token tiles with V_WMMA_F32_16X16X4_F32.
// Each wave holds B fragments in registers and reuses them across MT=4
// A-tiles (4x less weight traffic than a 16-token tile).
// TANH_RES: y = tanh(ffn + x) stored to y_out (stage A)
// READOUT : y = ffn + x, then out[t] = dot(y, wout) + bout (stage B)
// ---------------------------------------------------------------------------
template <bool TANH_RES, bool READOUT>
__global__ __launch_bounds__(256) void ffn_kernel(
    const float* __restrict__ x_in, const float* __restrict__ rms_w,
    const float* __restrict__ w1, const float* __restrict__ w2,
    const float* __restrict__ w3, float* __restrict__ y_out,
    const float* __restrict__ wout, const float* __restrict__ bout,
    float* __restrict__ scalar_out) {
  __shared__ float Xs[MT * 16][132];  // 132 % 64 == 4 -> conflict-free rows
  __shared__ float Gs[MT * 16][388];  // 388 % 64 == 4
  __shared__ float red[MT * 16][4];
  __shared__ float scale_s[MT * 16];

  const int tid = threadIdx.x;
  const size_t base = (size_t)blockIdx.x * (MT * 16) * DIM;

  // ---- Phase 0: load 64 tokens, RMSNorm into Xs ----
  {
    const int tok = tid >> 2;  // 0..63
    const int sub = tid & 3;   // 0..3, 32 elems each
    float ss = 0.f;
#pragma unroll
    for (int i = 0; i < 32; ++i) {
      float x = x_in[base + (size_t)tok * DIM + sub * 32 + i];
      Xs[tok][sub * 32 + i] = x;
      ss += x * x;
    }
    red[tok][sub] = ss;
    __syncthreads();
    if (tid < MT * 16) {
      float s = red[tid][0] + red[tid][1] + red[tid][2] + red[tid][3];
      scale_s[tid] = rsqrtf(s * (1.0f / DIM) + EPSF);
    }
    __syncthreads();
    const float sc = scale_s[tok];
#pragma unroll
    for (int i = 0; i < 32; ++i) {
      int d = sub * 32 + i;
      Xs[tok][d] = Xs[tok][d] * sc * rms_w[d];
    }
    __syncthreads();
  }

  const int wave = tid >> 5;  // 0..7
  const int lane = tid & 31;
  const int hl = lane >> 4;   // lane half (K split)
  const int ln = lane & 15;   // M for A, N for B/D

  // ---- Phase 1: H = silu(X@w1) * (X@w3); 24 N-chunks, 3 per wave ----
  for (int j = 0; j < 3; ++j) {
    const int n0 = (wave * 3 + j) * 16;
    v8f acc1[MT] = {};
    v8f acc3[MT] = {};
#pragma unroll 2
    for (int k0 = 0; k0 < DIM; k0 += 4) {
      const int k = k0 + 2 * hl;
      v2f b1; b1.x = w1[(size_t)k * HIDDEN + n0 + ln]; b1.y = w1[(size_t)(k + 1) * HIDDEN + n0 + ln];
      v2f b3; b3.x = w3[(size_t)k * HIDDEN + n0 + ln]; b3.y = w3[(size_t)(k + 1) * HIDDEN + n0 + ln];
#pragma unroll
      for (int mt = 0; mt < MT; ++mt) {
        v2f a; a.x = Xs[mt * 16 + ln][k]; a.y = Xs[mt * 16 + ln][k + 1];
        acc1[mt] = wmma4(a, b1, acc1[mt]);
        acc3[mt] = wmma4(a, b3, acc3[mt]);
      }
    }
#pragma unroll
    for (int mt = 0; mt < MT; ++mt) {
#pragma unroll
      for (int r = 0; r < 8; ++r) {
        float h1 = acc1[mt][r], h3 = acc3[mt][r];
        Gs[mt * 16 + r + 8 * hl][n0 + ln] = h1 * h3 * fast_sigmoid(h1);
      }
    }
  }
  __syncthreads();

  // ---- Phase 2: Y = H @ w2 (K = 384); one 16-col chunk per wave ----
  {
    const int n0 = wave * 16;
    v8f acc[MT] = {};
#pragma unroll 2
    for (int k0 = 0; k0 < HIDDEN; k0 += 4) {
      const int k = k0 + 2 * hl;
      v2f b; b.x = w2[(size_t)k * DIM + n0 + ln]; b.y = w2[(size_t)(k + 1) * DIM + n0 + ln];
#pragma unroll
      for (int mt = 0; mt < MT; ++mt) {
        v2f a; a.x = Gs[mt * 16 + ln][k]; a.y = Gs[mt * 16 + ln][k + 1];
        acc[mt] = wmma4(a, b, acc[mt]);
      }
    }
#pragma unroll
    for (int mt = 0; mt < MT; ++mt) {
#pragma unroll
      for (int r = 0; r < 8; ++r) {
        const int m = mt * 16 + r + 8 * hl;
        const int col = n0 + ln;
        const size_t idx = base + (size_t)m * DIM + col;
        float res = acc[mt][r] + x_in[idx];  // residual uses raw input
        if (TANH_RES) {
          y_out[idx] = fast_tanh(res);
        } else {
          Xs[m][col] = res;  // Xs dead after phase-1 barrier; reuse as Ys
        }
      }
    }
  }

  if (READOUT) {
    __syncthreads();
    if (tid < MT * 16) {
      float s = bout[0];
#pragma unroll 8
      for (int d = 0; d < DIM; ++d) s += Xs[tid][d] * wout[d];
      scalar_out[blockIdx.x * (MT * 16) + tid] = s;
    }
  }
}

// ---------------------------------------------------------------------------
// CSR build for the segment product
// ---------------------------------------------------------------------------
__global__ void zero_kernel(int* __restrict__ counts) {
  int i = blockIdx.x * blockDim.x + threadIdx.x;
  if (i < N_LOGICAL) counts[i] = 0;
}

__global__ void hist_kernel(const int* __restrict__ dst, int* __restrict__ counts) {
  int e = blockIdx.x * blockDim.x + threadIdx.x;
  if (e < EDGES) atomicAdd(&counts[dst[e]], 1);
}

__global__ void scan_kernel(const int* __restrict__ counts,
                            int* __restrict__ offsets, int* __restrict__ cursor) {
  __shared__ int partial[256];
  const int tid = threadIdx.x;
  const int base = tid * 16;  // 256 * 16 == 4096
  int s = 0;
#pragma unroll
  for (int i = 0; i < 16; ++i) s += counts[base + i];
  partial[tid] = s;
  __syncthreads();
  if (tid == 0) {
    int run = 0;
    for (int i = 0; i < 256; ++i) { int t = partial[i]; partial[i] = run; run += t; }
    offsets[N_LOGICAL] = run;
  }
  __syncthreads();
  int run = partial[tid];
#pragma unroll
  for (int i = 0; i < 16; ++i) {
    offsets[base + i] = run;
    cursor[base + i] = run;
    run += counts[base + i];
  }
}

__global__ void scatter_kernel(const int* __restrict__ dst, int* __restrict__ cursor,
                               int* __restrict__ edge_idx) {
  int e = blockIdx.x * blockDim.x + threadIdx.x;
  if (e < EDGES) {
    int pos = atomicAdd(&cursor[dst[e]], 1);
    edge_idx[pos] = e;
  }
}

// ---------------------------------------------------------------------------
// Segment product: block = (segment, batch); thread owns one of 128 dims.
// Gather indices staged through LDS in chunks of 128 (coalesced), then each
// edge iteration is one coalesced 512B row read.
// ---------------------------------------------------------------------------
__global__ __launch_bounds__(128) void segprod_kernel(
    const float* __restrict__ v_act, const int* __restrict__ data_src,
    const int* __restrict__ edge_idx, const int* __restrict__ offsets,
    float* __restrict__ l) {
  __shared__ int s_src[128];
  const int seg = blockIdx.x;
  const int b = blockIdx.y;
  const int d = threadIdx.x;
  const int s0 = offsets[seg];
  const int s1 = offsets[seg + 1];
  const float* vb = v_act + (size_t)b * N_DATA * DIM;
  float acc = 1.0f;
  for (int cs = s0; cs < s1; cs += 128) {
    const int n = min(128, s1 - cs);
    if (d < n) s_src[d] = data_src[edge_idx[cs + d]];
    __syncthreads();
    for (int i = 0; i < n; ++i) acc *= vb[(size_t)s_src[i] * DIM + d];
    __syncthreads();
  }
  l[((size_t)b * N_LOGICAL + seg) * DIM + d] = acc;
}

// ---------------------------------------------------------------------------
extern "C" void kernel_launch(void* const* d_in, const int* in_sizes, int n_in,
                              void* d_out, int out_size, void* d_ws, size_t ws_size,
                              hipStream_t stream) {
  const float* v        = (const float*)d_in[0];
  const int*   data_src = (const int*)d_in[1];
  const int*   data_dst = (const int*)d_in[2];
  // d_in[3] = num_logical scalar (compile-time constant here)
  const float* rms1_w = (const float*)d_in[4];
  const float* w1a    = (const float*)d_in[5];
  const float* w2a    = (const float*)d_in[6];
  const float* w3a    = (const float*)d_in[7];
  const float* rms2_w = (const float*)d_in[8];
  const float* w1b    = (const float*)d_in[9];
  const float* w2b    = (const float*)d_in[10];
  const float* w3b    = (const float*)d_in[11];
  const float* wout   = (const float*)d_in[12];
  const float* bout   = (const float*)d_in[13];

  char* ws = (char*)d_ws;
  size_t off = 0;
  auto carve = [&](size_t bytes) -> char* {
    char* p = ws + off;
    off = (off + bytes + 255) & ~(size_t)255;
    return p;
  };
  float* v_act   = (float*)carve((size_t)BB * N_DATA * DIM * sizeof(float));
  float* l_buf   = (float*)carve((size_t)BB * N_LOGICAL * DIM * sizeof(float));
  int*   counts  = (int*)carve(N_LOGICAL * sizeof(int));
  int*   offsets = (int*)carve((N_LOGICAL + 1) * sizeof(int));
  int*   cursor  = (int*)carve(N_LOGICAL * sizeof(int));
  int*   edge_ix = (int*)carve(EDGES * sizeof(int));

  // Stage A: v_act = tanh(ffn_a(rmsnorm(v)) + v) over 400k tokens
  const int ntilesA = (BB * N_DATA) / (MT * 16);  // 6250
  ffn_kernel<true, false><<<ntilesA, 256, 0, stream>>>(
      v, rms1_w, w1a, w2a, w3a, v_act, nullptr, nullptr, nullptr);

  // CSR of data_dst
  zero_kernel<<<(N_LOGICAL + 255) / 256, 256, 0, stream>>>(counts);
  hist_kernel<<<(EDGES + 255) / 256, 256, 0, stream>>>(data_dst, counts);
  scan_kernel<<<1, 256, 0, stream>>>(counts, offsets, cursor);
  scatter_kernel<<<(EDGES + 255) / 256, 256, 0, stream>>>(data_dst, cursor, edge_ix);

  // Segment product into l
  dim3 gseg(N_LOGICAL, BB);
  segprod_kernel<<<gseg, 128, 0, stream>>>(v_act, data_src, edge_ix, offsets, l_buf);

  // Stage B: ffn_b + residual + readout -> (B, N_LOGICAL)
  const int ntilesB = (BB * N_LOGICAL) / (MT * 16);  // 256
  ffn_kernel<false, true><<<ntilesB, 256, 0, stream>>>(
      l_buf, rms2_w, w1b, w2b, w3b, nullptr, wout, bout, (float*)d_out);
}